// GNN_6554120093878
// MI455X (gfx1250) — compile-verified
//
#include <hip/hip_runtime.h>

typedef float v2f __attribute__((ext_vector_type(2)));
typedef float v8f __attribute__((ext_vector_type(8)));

#define N_NODES  100000
#define N_EDGES  1600000
#define N_GRAPHS 256
#define IN_DIM   128
#define GCN_DIM  128
#define FCN_DIM  256
#define OUT_DIM  8

// ---------------------------------------------------------------- utilities

__global__ void zero_kernel(float* __restrict__ p, long long n) {
  long long i = (long long)blockIdx.x * blockDim.x + threadIdx.x;
  long long stride = (long long)gridDim.x * blockDim.x;
  for (; i < n; i += stride) p[i] = 0.0f;
}

__global__ void degree_kernel(const int* __restrict__ src, const int* __restrict__ dst,
                              float* __restrict__ outdeg, float* __restrict__ indeg, int n) {
  int i = blockIdx.x * blockDim.x + threadIdx.x;
  if (i < n) {
    unsafeAtomicAdd(&outdeg[src[i]], 1.0f);
    unsafeAtomicAdd(&indeg[dst[i]], 1.0f);
  }
}

__global__ void norm_kernel(const float* __restrict__ outdeg, const float* __restrict__ indeg,
                            float* __restrict__ nsrc, float* __restrict__ ndst, int n) {
  int i = blockIdx.x * blockDim.x + threadIdx.x;
  if (i < n) {
    nsrc[i] = rsqrtf(fmaxf(outdeg[i], 1.0f));
    ndst[i] = rsqrtf(fmaxf(indeg[i], 1.0f));
  }
}

// ------------------------------------------------------- WMMA f32 GEMM
// One wave computes one 16x16 tile via V_WMMA_F32_16X16X4_F32.
// Block = 8 waves (32x8) sharing a 16xK A-tile in LDS (padded rows, no bank
// conflicts). K, N compile-time. Epilogue mode EPI is compile-time:
//   EPI==0 : C = (A@W) * rowscale[row]          (GCN linear, pre-scatter)
//   EPI==1 : C = relu(A@W + bias[col])          (MLP hidden)
//   EPI==2 : C = A@W + bias[col]                (MLP out)
// Column guard only instantiated when N % 16 != 0 (branch-free cndmask).
template <int K, int N, int EPI>
__global__ __launch_bounds__(256) void gemm_wmma_f32(
    const float* __restrict__ A, const float* __restrict__ W,
    const float* __restrict__ rowscale, const float* __restrict__ bias,
    float* __restrict__ C) {
  constexpr bool GUARD = (N % 16) != 0;
  constexpr int NT = (N + 15) / 16;      // N tiles
  constexpr int LDK = K + 4;             // padded LDS row stride (floats)
  __shared__ float ldsA[16 * LDK];

  const int tid = threadIdx.y * 32 + threadIdx.x;
  const long long rowBase = (long long)blockIdx.x * 16;

  // Stage A tile: 16 consecutive rows = one contiguous 16*K block. float4 copy.
  {
    const float4* __restrict__ src4 = (const float4*)(A + rowBase * K);
    #pragma unroll
    for (int j = tid; j < (16 * K) / 4; j += 256) {
      const int r  = j / (K / 4);
      const int cj = j - r * (K / 4);
      float4 v = src4[j];
      float* d = &ldsA[r * LDK + cj * 4];
      d[0] = v.x; d[1] = v.y; d[2] = v.z; d[3] = v.w;
    }
  }
  __syncthreads();

  const int tileN = blockIdx.y * 8 + threadIdx.y;
  if (tileN >= NT) return;  // wave-uniform; EXEC stays all-ones for WMMA

  const int lane = threadIdx.x;   // 0..31
  const int half = lane >> 4;     // 0: K pair {k,k+1}, 1: K pair {k+2,k+3}
  const int l    = lane & 15;
  const int col  = tileN * 16 + l;
  const int colc = GUARD ? (col < N ? col : N - 1) : col;   // clamped, branch-free
  const bool cok = GUARD ? (col < N) : true;

  v8f acc = {0.f, 0.f, 0.f, 0.f, 0.f, 0.f, 0.f, 0.f};
  const float* __restrict__ ar = &ldsA[l * LDK + half * 2];  // A[l][k + 2*half]
  const float* __restrict__ wp = W + colc;                   // column pointer

  #pragma unroll 4
  for (int k = 0; k < K; k += 4) {
    v2f a;
    a.x = ar[k];
    a.y = ar[k + 1];
    const int kb = k + half * 2;
    float w0 = wp[(long long)kb * N];
    float w1 = wp[(long long)(kb + 1) * N];
    v2f b;
    b.x = cok ? w0 : 0.0f;   // folds away when !GUARD
    b.y = cok ? w1 : 0.0f;
    acc = __builtin_amdgcn_wmma_f32_16x16x4_f32(
        /*neg_a=*/false, a, /*neg_b=*/false, b,
        /*c_mod=*/(short)0, acc, /*reuse_a=*/false, /*reuse_b=*/false);
  }

  // ---- epilogue (compile-time specialized, vectorized rowscale fetch) ----
  float rs[8];
  if constexpr (EPI == 0) {
    // 8 consecutive rows for this lane's half: two float4 loads (broadcast)
    const float4* __restrict__ rp = (const float4*)(rowscale + rowBase + half * 8);
    float4 r0 = rp[0], r1 = rp[1];
    rs[0] = r0.x; rs[1] = r0.y; rs[2] = r0.z; rs[3] = r0.w;
    rs[4] = r1.x; rs[5] = r1.y; rs[6] = r1.z; rs[7] = r1.w;
  }
  float bi = 0.0f;
  if constexpr (EPI >= 1) bi = bias[colc];

  if (cok) {
    #pragma unroll
    for (int r = 0; r < 8; ++r) {
      const long long row = rowBase + half * 8 + r;  // C/D layout: M = r + 8*half
      float v = acc[r];
      if constexpr (EPI == 0) v *= rs[r];
      if constexpr (EPI >= 1) v += bi;
      if constexpr (EPI == 1) v = fmaxf(v, 0.0f);
      C[row * N + col] = v;
    }
  }
}

// ------------------------------------------------------- edge scatter-add
// One wave per edge: 32 lanes x float4 = the full 128-dim row.
__global__ void scatter_kernel(const float* __restrict__ hlin, const int* __restrict__ src,
                               const int* __restrict__ dst, float* __restrict__ agg, int n_edges) {
  long long gtid = (long long)blockIdx.x * blockDim.x + threadIdx.x;
  int e = (int)(gtid >> 5);
  int lane = threadIdx.x & 31;
  if (e >= n_edges) return;
  int s = src[e];
  int d = dst[e];
  const float4 v = ((const float4*)(hlin + (long long)s * GCN_DIM))[lane];
  float* out = agg + (long long)d * GCN_DIM + lane * 4;
  unsafeAtomicAdd(out + 0, v.x);
  unsafeAtomicAdd(out + 1, v.y);
  unsafeAtomicAdd(out + 2, v.z);
  unsafeAtomicAdd(out + 3, v.w);
}

// h[i] = relu(agg[i] * norm_dst[node] + bias[dim])
__global__ void finalize_kernel(const float* __restrict__ agg, const float* __restrict__ ndst,
                                const float* __restrict__ bias, float* __restrict__ h,
                                long long n_elems) {
  long long i = (long long)blockIdx.x * blockDim.x + threadIdx.x;
  if (i < n_elems) {
    int node = (int)(i >> 7);       // /128
    int d = (int)(i & 127);
    h[i] = fmaxf(agg[i] * ndst[node] + bias[d], 0.0f);
  }
}

// Per-graph feature sums + counts. One wave per node.
__global__ void readout_sum_kernel(const float* __restrict__ h, const int* __restrict__ n2g,
                                   float* __restrict__ sums, float* __restrict__ cnt, int n_nodes) {
  long long gtid = (long long)blockIdx.x * blockDim.x + threadIdx.x;
  int node = (int)(gtid >> 5);
  int lane = threadIdx.x & 31;
  if (node >= n_nodes) return;
  int g = n2g[node];
  const float4 v = ((const float4*)(h + (long long)node * GCN_DIM))[lane];
  float* out = sums + (long long)g * GCN_DIM + lane * 4;
  unsafeAtomicAdd(out + 0, v.x);
  unsafeAtomicAdd(out + 1, v.y);
  unsafeAtomicAdd(out + 2, v.z);
  unsafeAtomicAdd(out + 3, v.w);
  if (lane == 0) unsafeAtomicAdd(&cnt[g], 1.0f);
}

__global__ void readout_div_kernel(float* __restrict__ sums, const float* __restrict__ cnt) {
  int i = blockIdx.x * blockDim.x + threadIdx.x;
  if (i < N_GRAPHS * GCN_DIM) sums[i] /= fmaxf(cnt[i >> 7], 1.0f);
}

// ---------------------------------------------------------------- launcher

extern "C" void kernel_launch(void* const* d_in, const int* in_sizes, int n_in,
                              void* d_out, int out_size, void* d_ws, size_t ws_size,
                              hipStream_t stream) {
  const float* x    = (const float*)d_in[0];
  const int* esrc   = (const int*)d_in[1];
  const int* edst   = (const int*)d_in[2];
  const int* n2g    = (const int*)d_in[3];
  const float* Wg0  = (const float*)d_in[4];
  const float* bg0  = (const float*)d_in[5];
  const float* Wg1  = (const float*)d_in[6];
  const float* bg1  = (const float*)d_in[7];
  const float* Wg2  = (const float*)d_in[8];
  const float* bg2  = (const float*)d_in[9];
  const float* Wf0  = (const float*)d_in[10];
  const float* bf0  = (const float*)d_in[11];
  const float* Wf1  = (const float*)d_in[12];
  const float* bf1  = (const float*)d_in[13];
  const float* Wout = (const float*)d_in[14];
  const float* bout = (const float*)d_in[15];
  float* out = (float*)d_out;

  // Workspace partition (floats)
  float* ws = (float*)d_ws;
  size_t off = 0;
  float* outdeg = ws + off; off += N_NODES;
  float* indeg  = ws + off; off += N_NODES;
  float* nsrc   = ws + off; off += N_NODES;
  float* ndst   = ws + off; off += N_NODES;
  float* gsum   = ws + off; off += (size_t)N_GRAPHS * GCN_DIM;
  float* gcnt   = ws + off; off += N_GRAPHS;
  float* fc1    = ws + off; off += (size_t)N_GRAPHS * FCN_DIM;
  float* fc2    = ws + off; off += (size_t)N_GRAPHS * FCN_DIM;
  float* buf0   = ws + off; off += (size_t)N_NODES * GCN_DIM;  // GEMM out (h*W*norm_src)
  float* buf1   = ws + off; off += (size_t)N_NODES * GCN_DIM;  // scatter accumulator
  float* buf2   = ws + off; off += (size_t)N_NODES * GCN_DIM;  // layer activations

  const long long NH = (long long)N_NODES * GCN_DIM;
  const dim3 gemmBlk(32, 8, 1);

  // Degrees & norms
  zero_kernel<<<4096, 256, 0, stream>>>(outdeg, 2LL * N_NODES);  // outdeg+indeg contiguous
  degree_kernel<<<(N_EDGES + 255) / 256, 256, 0, stream>>>(esrc, edst, outdeg, indeg, N_EDGES);
  norm_kernel<<<(N_NODES + 255) / 256, 256, 0, stream>>>(outdeg, indeg, nsrc, ndst, N_NODES);

  const float* Wg[3] = {Wg0, Wg1, Wg2};
  const float* bg[3] = {bg0, bg1, bg2};

  // 3 GCN layers
  for (int layer = 0; layer < 3; ++layer) {
    const float* hin = (layer == 0) ? x : buf2;
    gemm_wmma_f32<GCN_DIM, GCN_DIM, 0><<<dim3(N_NODES / 16, 1), gemmBlk, 0, stream>>>(
        hin, Wg[layer], nsrc, nullptr, buf0);
    zero_kernel<<<8192, 256, 0, stream>>>(buf1, NH);
    scatter_kernel<<<(int)(((long long)N_EDGES * 32 + 255) / 256), 256, 0, stream>>>(
        buf0, esrc, edst, buf1, N_EDGES);
    finalize_kernel<<<(int)((NH + 255) / 256), 256, 0, stream>>>(buf1, ndst, bg[layer], buf2, NH);
  }

  // Graph mean readout
  zero_kernel<<<64, 256, 0, stream>>>(gsum, (long long)N_GRAPHS * GCN_DIM + N_GRAPHS);
  readout_sum_kernel<<<(int)(((long long)N_NODES * 32 + 255) / 256), 256, 0, stream>>>(
      buf2, n2g, gsum, gcnt, N_NODES);
  readout_div_kernel<<<(N_GRAPHS * GCN_DIM + 255) / 256, 256, 0, stream>>>(gsum, gcnt);

  // MLP head (WMMA too)
  gemm_wmma_f32<GCN_DIM, FCN_DIM, 1><<<dim3(N_GRAPHS / 16, 2), gemmBlk, 0, stream>>>(
      gsum, Wf0, nullptr, bf0, fc1);
  gemm_wmma_f32<FCN_DIM, FCN_DIM, 1><<<dim3(N_GRAPHS / 16, 2), gemmBlk, 0, stream>>>(
      fc1, Wf1, nullptr, bf1, fc2);
  gemm_wmma_f32<FCN_DIM, OUT_DIM, 2><<<dim3(N_GRAPHS / 16, 1), gemmBlk, 0, stream>>>(
      fc2, Wout, nullptr, bout, out);
}